// BoxDecoder_15169824489590
// MI455X (gfx1250) — compile-verified
//
#include <hip/hip_runtime.h>
#include <hip/hip_bf16.h>
#include <stdint.h>

// BoxDecoder for MI455X (gfx1250), compile-only tuned.
//
// Roofline: ~78 MB read + ~410 MB written per call -> pure HBM stream,
// ~21 us floor at 23.3 TB/s. No matrix ops exist in this workload, so WMMA is
// inapplicable; instead we exercise the CDNA5 async data-movement path:
// rows are staged in a double-buffered LDS tile and pushed to memory with
// GLOBAL_STORE_ASYNC_FROM_LDS_B128 (tracked by ASYNCcnt), giving fully
// coalesced 512B-per-wave-instruction stores overlapped with the next batch
// iteration's compute. Small side outputs (mask, batch_index) use
// nontemporal b128 stores.

typedef float v4f __attribute__((ext_vector_type(4)));

#define TPB 128                      // 4 wave32s per block
#define V4_PER_THREAD 12             // 8 rows * 6 floats = 48 floats = 12 float4
#define BUF_V4 (TPB * V4_PER_THREAD) // 1536 float4 = 24 KB per buffer
#define GRID_Y 4                     // split the 16 batch items 4 ways

__global__ __launch_bounds__(TPB)
void box_decode_async(const float* __restrict__ anchors,
                      const float* __restrict__ loc,
                      const float* __restrict__ cls,
                      const float* __restrict__ pvarx,
                      const float* __restrict__ pvary,
                      const float* __restrict__ pthresh,
                      float* __restrict__ out,
                      int A, int B)
{
    __shared__ v4f smem[2 * BUF_V4];   // 48 KB double-buffered staging tile

    const int tid   = threadIdx.x;
    const int base  = blockIdx.x * TPB;
    const int a     = base + tid;
    const bool valid = (a < A);
    const int nvalid = (A - base) < TPB ? (A - base) : TPB;
    const int cnt    = nvalid * V4_PER_THREAD;   // float4s in this block's tile

    // Scalar params live in device memory (graph-capture safe): uniform loads.
    const float varx = *pvarx;
    const float vary = *pvary;
    const float thr  = *pthresh;

    const size_t N = (size_t)B * (size_t)A * 8;  // B*A*C rows
    float* out_box   = out;           // [N,6]
    float* out_mask  = out + N * 6;   // [N]
    float* out_batch = out + N * 7;   // [N]

    // Anchor loaded ONCE, reused across all batch items handled by this block.
    v4f anc;
    if (valid) {
        anc = ((const v4f*)anchors)[a];          // (cx, cy, w, h)
    } else {
        anc.x = 0.f; anc.y = 0.f; anc.z = 1.f; anc.w = 1.f;
    }

    // Low 32 bits of a generic pointer to LDS == LDS byte offset on amdgcn.
    const uint32_t lds_base = (uint32_t)(uintptr_t)(&smem[0]);

    const int bpg = (B + GRID_Y - 1) / GRID_Y;
    const int b0  = (int)blockIdx.y * bpg;
    const int bN  = (b0 + bpg) < B ? (b0 + bpg) : B;

    for (int b = b0; b < bN; ++b) {
        const int buf = b & 1;

        // WAR protection for the LDS buffer we are about to overwrite:
        // each wave issues <=12 async stores per iteration and ASYNCcnt is
        // in-order, so <=12 outstanding means the stores from two iterations
        // ago (same buffer) are complete. The barrier extends that guarantee
        // across all waves reading this buffer.
        asm volatile("s_wait_asynccnt 12" ::: "memory");
        __syncthreads();

        if (valid) {
            const size_t ba = (size_t)b * (size_t)A + (size_t)a;

            v4f l = ((const v4f*)loc)[ba];       // (dx, dy, dw, dh)
            float x  = l.x * varx * anc.z + anc.x;
            float y  = l.y * varx * anc.w + anc.y;
            float w  = __expf(l.z * vary) * anc.z;
            float h  = __expf(l.w * vary) * anc.w;
            float x0 = x - 0.5f * w, y0 = y - 0.5f * h;
            float x1 = x + 0.5f * w, y1 = y + 0.5f * h;

            v4f s0 = ((const v4f*)cls)[ba * 2];
            v4f s1 = ((const v4f*)cls)[ba * 2 + 1];
            float sc[8] = {s0.x, s0.y, s0.z, s0.w, s1.x, s1.y, s1.z, s1.w};

            // Build 8 rows of 6 floats (192 B) in registers, then 12x b128
            // DS stores into the staging tile (tid-major layout).
            float rows[48];
#pragma unroll
            for (int c = 0; c < 8; ++c) {
                rows[c * 6 + 0] = x0;
                rows[c * 6 + 1] = y0;
                rows[c * 6 + 2] = x1;
                rows[c * 6 + 3] = y1;
                rows[c * 6 + 4] = sc[c];
                rows[c * 6 + 5] = (float)(c + 1);
            }
            v4f* dst = &smem[buf * BUF_V4 + tid * V4_PER_THREAD];
#pragma unroll
            for (int k = 0; k < V4_PER_THREAD; ++k)
                dst[k] = ((const v4f*)rows)[k];

            // mask (0/1 as float) and batch_index (as float): direct
            // nontemporal b128 stores (streamed once, never re-read here).
            v4f m0, m1, bi;
            m0.x = sc[0] >= thr ? 1.f : 0.f;  m0.y = sc[1] >= thr ? 1.f : 0.f;
            m0.z = sc[2] >= thr ? 1.f : 0.f;  m0.w = sc[3] >= thr ? 1.f : 0.f;
            m1.x = sc[4] >= thr ? 1.f : 0.f;  m1.y = sc[5] >= thr ? 1.f : 0.f;
            m1.z = sc[6] >= thr ? 1.f : 0.f;  m1.w = sc[7] >= thr ? 1.f : 0.f;
            bi.x = bi.y = bi.z = bi.w = (float)b;

            const size_t mi = ba * 8;            // 32 B aligned
            __builtin_nontemporal_store(m0, (v4f*)(out_mask + mi));
            __builtin_nontemporal_store(m1, (v4f*)(out_mask + mi + 4));
            __builtin_nontemporal_store(bi, (v4f*)(out_batch + mi));
            __builtin_nontemporal_store(bi, (v4f*)(out_batch + mi + 4));
        }

        // Make this buffer's DS writes visible before the DMA reads them.
        __syncthreads();

        // Stream the 24 KB tile (nvalid*192 B) to global memory with the
        // CDNA5 async LDS->global path. Per wave instruction: 32 lanes x 16 B
        // fully contiguous. Byte offsets fit in 32 bits (max ~307 MB).
        const uint32_t tile = (uint32_t)(((uint64_t)b * (uint64_t)A +
                                          (uint64_t)base) * 192ull);
        const uint32_t lbuf = lds_base + (uint32_t)(buf * BUF_V4 * 16);
        for (int i = tid; i < cnt; i += TPB) {
            const uint32_t g  = tile + (uint32_t)i * 16u;
            const uint32_t ls = lbuf + (uint32_t)i * 16u;
            asm volatile("global_store_async_from_lds_b128 %0, %1, %2"
                         :: "v"(g), "v"(ls), "s"(out_box)
                         : "memory");
        }
    }
    // S_ENDPGM performs an implicit wait-idle, draining remaining ASYNCcnt.
}

extern "C" void kernel_launch(void* const* d_in, const int* in_sizes, int n_in,
                              void* d_out, int out_size, void* d_ws, size_t ws_size,
                              hipStream_t stream) {
    (void)n_in; (void)d_ws; (void)ws_size; (void)out_size;

    const float* anchors = (const float*)d_in[0];  // [A,4]
    const float* loc     = (const float*)d_in[1];  // [B,A,4]
    const float* cls     = (const float*)d_in[2];  // [B,A,8]
    const float* pvarx   = (const float*)d_in[3];
    const float* pvary   = (const float*)d_in[4];
    const float* pthresh = (const float*)d_in[5];
    // d_in[6] = batch_size (unused by the dense outputs)

    const int A = in_sizes[0] / 4;
    const int B = in_sizes[1] / (A * 4);

    const int gx = (A + TPB - 1) / TPB;
    hipLaunchKernelGGL(box_decode_async, dim3(gx, GRID_Y, 1), dim3(TPB, 1, 1),
                       0, stream,
                       anchors, loc, cls, pvarx, pvary, pthresh,
                       (float*)d_out, A, B);
}